// Attention_81750407512242
// MI455X (gfx1250) — compile-verified
//
#include <hip/hip_runtime.h>
#include <math.h>

// Attention pooling, S=8192 B=32 H=256, fp32.
// Single-pass online-softmax (flash style): enc_output (268 MB) is streamed
// from HBM exactly once (second use of each tile hits WGP$), so runtime floor
// ~= 268MB / 23.3TB/s ~= 11.5us. Scores are computed with
// V_WMMA_F32_16X16X4_F32 (16 rows x K=256 per wave) which avoids all
// cross-lane reductions; hid is staged in LDS.

#define S_TOT   8192
#define BATCH   32
#define HDIM    256
#define CHUNK   256                   // S-rows per workgroup
#define NCHUNK  (S_TOT / CHUNK)       // 32 chunks
#define NWAVE   8                     // 256 threads, wave32
#define PSTRIDE 264                   // floats per per-chunk partial record

typedef __attribute__((ext_vector_type(2))) float v2f;
typedef __attribute__((ext_vector_type(8))) float v8f;

__global__ __launch_bounds__(256) void attn_pass1(
    const float* __restrict__ enc,   // [S, B, H]
    const float* __restrict__ hid,   // [1, B, H]
    float* __restrict__ ws)          // [B][NCHUNK][PSTRIDE]
{
    const int chunk = blockIdx.x;
    const int b     = blockIdx.y;
    const int tid   = threadIdx.x;
    const int lane  = tid & 31;
    const int wave  = tid >> 5;

    __shared__ __align__(16) float s_hid[HDIM];
    __shared__ float s_m[NWAVE];
    __shared__ float s_l[NWAVE];
    __shared__ __align__(16) float s_acc[NWAVE][HDIM];

    s_hid[tid] = hid[b * HDIM + tid];
    __syncthreads();

    // WMMA f32 16x16x4 A layout: VGPR0 = K0 (lanes 0-15) / K2 (lanes 16-31),
    // VGPR1 = K1 / K3. So per step t each lane loads a contiguous float2 at
    // h = 4t + o2. B (hid, replicated across all 16 columns) uses the same
    // K striping.
    const int o2 = (lane >> 4) << 1;       // 0 for low half, 2 for high half
    const int r0 = lane & 15;              // row within the 16-row tile

    float m = -INFINITY;                   // online softmax running max
    float l = 0.0f;                        // running sum of exp
    float acc[8] = {0, 0, 0, 0, 0, 0, 0, 0};  // lane owns h = 8*lane..8*lane+7

    for (int tile = wave; tile < CHUNK / 16; tile += NWAVE) {
        const int s0 = chunk * CHUNK + tile * 16;

        // ---- scores for 16 rows: D += A(16x4) * B(4x16), K = 256 in 64 steps
        v8f d = {0, 0, 0, 0, 0, 0, 0, 0};
        const float* arow =
            enc + ((size_t)(s0 + r0) * BATCH + b) * HDIM + o2;
        #pragma unroll 8
        for (int t = 0; t < HDIM / 4; ++t) {
            float2 a2 = *(const float2*)(arow + 4 * t);
            float2 h2 = *(const float2*)(&s_hid[4 * t + o2]);
            v2f av; av.x = a2.x; av.y = a2.y;
            v2f bv; bv.x = h2.x; bv.y = h2.y;
            d = __builtin_amdgcn_wmma_f32_16x16x4_f32(
                    false, av, false, bv, (short)0, d, false, false);
        }

        // D layout: VGPR j -> M=j (lanes 0-15) / M=8+j (lanes 16-31); all N
        // columns identical (B columns replicated). Broadcast 16 scores.
        float sc[16];
        #pragma unroll
        for (int j = 0; j < 8; ++j) {
            sc[j] = __uint_as_float(
                __builtin_amdgcn_readlane(__float_as_uint(d[j]), 0));
            sc[j + 8] = __uint_as_float(
                __builtin_amdgcn_readlane(__float_as_uint(d[j]), 16));
        }

        // ---- online softmax update over these 16 rows
        float mt = sc[0];
        #pragma unroll
        for (int r = 1; r < 16; ++r) mt = fmaxf(mt, sc[r]);
        float mnew  = fmaxf(m, mt);
        float scale = __expf(m - mnew);    // first tile: exp(-inf)=0
        l *= scale;
        #pragma unroll
        for (int k = 0; k < 8; ++k) acc[k] *= scale;

        float e[16];
        #pragma unroll
        for (int r = 0; r < 16; ++r) {
            e[r] = __expf(sc[r] - mnew);
            l += e[r];
        }

        // ---- weighted accumulation; rows re-read hit WGP$/L0 (tile = 16KB)
        #pragma unroll
        for (int r = 0; r < 16; ++r) {
            const float4* rb = (const float4*)(
                enc + ((size_t)(s0 + r) * BATCH + b) * HDIM + 8 * lane);
            float4 x0 = rb[0];
            float4 x1 = rb[1];
            acc[0] = fmaf(e[r], x0.x, acc[0]);
            acc[1] = fmaf(e[r], x0.y, acc[1]);
            acc[2] = fmaf(e[r], x0.z, acc[2]);
            acc[3] = fmaf(e[r], x0.w, acc[3]);
            acc[4] = fmaf(e[r], x1.x, acc[4]);
            acc[5] = fmaf(e[r], x1.y, acc[5]);
            acc[6] = fmaf(e[r], x1.z, acc[6]);
            acc[7] = fmaf(e[r], x1.w, acc[7]);
        }
        m = mnew;
    }

    // ---- combine the 8 wave partials within the workgroup via LDS
    #pragma unroll
    for (int k = 0; k < 8; ++k) s_acc[wave][8 * lane + k] = acc[k];
    if (lane == 0) { s_m[wave] = m; s_l[wave] = l; }
    __syncthreads();

    float M = s_m[0];
    #pragma unroll
    for (int w = 1; w < NWAVE; ++w) M = fmaxf(M, s_m[w]);
    float L = 0.0f, c = 0.0f;
    #pragma unroll
    for (int w = 0; w < NWAVE; ++w) {
        float ew = __expf(s_m[w] - M);
        L += ew * s_l[w];
        c += ew * s_acc[w][tid];
    }

    float* rec = ws + ((size_t)b * NCHUNK + chunk) * PSTRIDE;
    rec[tid] = c;                       // unnormalized context partial
    if (tid == 0) rec[HDIM]     = M;
    if (tid == 1) rec[HDIM + 1] = L;
}

// Split-softmax merge across chunks: one block per batch, one thread per h.
__global__ __launch_bounds__(256) void attn_pass2(
    const float* __restrict__ ws, float* __restrict__ out)
{
    const int b = blockIdx.x;
    const int h = threadIdx.x;
    const float* base = ws + (size_t)b * NCHUNK * PSTRIDE;

    float M = -INFINITY;
    for (int c = 0; c < NCHUNK; ++c)
        M = fmaxf(M, base[c * PSTRIDE + HDIM]);

    float L = 0.0f, v = 0.0f;
    for (int c = 0; c < NCHUNK; ++c) {
        float ew = __expf(base[c * PSTRIDE + HDIM] - M);
        L += ew * base[c * PSTRIDE + HDIM + 1];
        v += ew * base[c * PSTRIDE + h];
    }
    out[b * HDIM + h] = v / L;
}

extern "C" void kernel_launch(void* const* d_in, const int* in_sizes, int n_in,
                              void* d_out, int out_size, void* d_ws, size_t ws_size,
                              hipStream_t stream) {
    const float* enc = (const float*)d_in[0];   // [S, B, H] fp32
    const float* hid = (const float*)d_in[1];   // [1, B, H] fp32
    float* out = (float*)d_out;                 // [B, H] fp32
    float* ws  = (float*)d_ws;                  // needs B*NCHUNK*PSTRIDE*4 ~= 1.03 MB

    dim3 grid1(NCHUNK, BATCH);
    attn_pass1<<<grid1, 256, 0, stream>>>(enc, hid, ws);
    attn_pass2<<<BATCH, 256, 0, stream>>>(ws, out);
}